// CTCLayer_73246372266002
// MI455X (gfx1250) — compile-verified
//
#include <hip/hip_runtime.h>
#include <stdint.h>

// CTC forward loss, MI455X (gfx1250, wave32).
// One wave per batch element; 257 alpha states in registers, 8 per lane
// (s = 8*lane + j) + state 256 as a uniform extra slot (valid on lane 31).
// Even states are blanks at compile time -> 1-exp lse2 + shared blank
// log-prob; odd states' classes are the lane's own 4 labels. Per step:
// 5 LDS gathers, 1 shuffle, 31 TRANS ops, all in log2 domain
// (v_exp_f32 / v_log_f32; recursion is scale-invariant, ln2 applied once
// at the end). A 16-deep global->LDS async pipeline (ASYNCcnt) streams
// y_pred rows; the scan loop is 2x software-pipelined under a single
// s_wait_asynccnt 14.

#define NEGF (-1.0e30f)
#define EPSF (1.0e-07f)

constexpr int B_    = 256;
constexpr int T_    = 1024;
constexpr int C_    = 128;
constexpr int U_    = 128;
constexpr int NS    = 16;        // async pipeline depth (stages)
constexpr int BLANK = C_ - 1;    // 127

__device__ __forceinline__ float fexp2(float x) { return __builtin_amdgcn_exp2f(x); }
__device__ __forceinline__ float flog2(float x) { return __builtin_amdgcn_logf(x); }

// lse2: the max-term exponential is exactly 1 -> 1 exp + 1 log.
__device__ __forceinline__ float lse2(float x0, float x1) {
  const float mx = fmaxf(x0, x1);
  const float mn = fminf(x0, x1);
  return mx + flog2(1.0f + fexp2(mn - mx));
}
// lse3: 3 parallel exps keep the dependency chain shallow (max3 -> exp -> log).
__device__ __forceinline__ float lse3(float x0, float x1, float x2) {
  const float m = fmaxf(fmaxf(x0, x1), x2);
  return m + flog2(fexp2(x0 - m) + fexp2(x1 - m) + fexp2(x2 - m));
}

// Copy 16 B/lane (512 B/wave = one y_pred row) global -> LDS, ASYNCcnt-tracked.
__device__ __forceinline__ void async_row_load(const float* gsrc, uint32_t lds_off) {
  asm volatile("global_load_async_to_lds_b128 %0, %1, off"
               :: "v"(lds_off), "v"((uint64_t)(uintptr_t)gsrc)
               : "memory");
}
__device__ __forceinline__ void wait_async_15() {
  asm volatile("s_wait_asynccnt 15" ::: "memory");
}
__device__ __forceinline__ void wait_async_14() {
  asm volatile("s_wait_asynccnt 14" ::: "memory");
}

// One alpha-recursion step against an LDS row stage.
__device__ __forceinline__ void ctc_step(float (&a)[9], const float* __restrict__ stg,
                                         const int y0, const int y1, const int y2, const int y3,
                                         const bool al0, const bool al1,
                                         const bool al2, const bool al3,
                                         const int lane) {
  // log2 prob gathers: 1 blank (wave-broadcast address) + 4 labels
  const float lpb = flog2(stg[BLANK] + EPSF);
  const float lp1 = flog2(stg[y0] + EPSF);
  const float lp3 = flog2(stg[y1] + EPSF);
  const float lp5 = flog2(stg[y2] + EPSF);
  const float lp7 = flog2(stg[y3] + EPSF);

  // boundary alpha (state s0-1) from the previous lane
  float am1 = __shfl_up(a[7], 1, 32);
  if (lane == 0) am1 = NEGF;

  const float n0 = lse2(a[0], am1)                     + lpb;  // s0   (blank)
  const float n1 = lse3(a[1], a[0], al0 ? am1  : NEGF) + lp1;  // s0+1 (label)
  const float n2 = lse2(a[2], a[1])                    + lpb;  // s0+2 (blank)
  const float n3 = lse3(a[3], a[2], al1 ? a[1] : NEGF) + lp3;  // s0+3
  const float n4 = lse2(a[4], a[3])                    + lpb;  // s0+4 (blank)
  const float n5 = lse3(a[5], a[4], al2 ? a[3] : NEGF) + lp5;  // s0+5
  const float n6 = lse2(a[6], a[5])                    + lpb;  // s0+6 (blank)
  const float n7 = lse3(a[7], a[6], al3 ? a[5] : NEGF) + lp7;  // s0+7
  const float n8 = lse2(a[8], a[7])                    + lpb;  // s0+8 (state 256 on lane 31)

  a[0] = n0; a[1] = n1; a[2] = n2; a[3] = n3; a[4] = n4;
  a[5] = n5; a[6] = n6; a[7] = n7; a[8] = n8;
}

__global__ __launch_bounds__(32)
void ctc_fwd_kernel(const int* __restrict__ y_true,
                    const float* __restrict__ y_pred,
                    float* __restrict__ out)
{
  __shared__ float rows[NS * C_];   // 16 stages x 128 floats = 8 KB

  const int b    = blockIdx.x;
  const int lane = threadIdx.x;     // 0..31, wave32
  const float* yb = y_pred + (size_t)b * T_ * C_;

  // Low 32 bits of a generic LDS pointer == DS byte offset.
  const uint32_t rowsOff = (uint32_t)(uintptr_t)(&rows[0]);
  const uint32_t laneB   = (uint32_t)(lane * 16);

  // ---- prologue: fill the 16-stage pipeline with rows 0..15 ----
  #pragma unroll
  for (int k = 0; k < NS; ++k)
    async_row_load(yb + k * C_ + lane * 4, rowsOff + (uint32_t)(k * C_ * 4) + laneB);

  // ---- labels: this lane's 4 odd-state classes (coalesced b128 load) ----
  const int4 yv = *(const int4*)(y_true + (size_t)b * U_ + lane * 4);
  const int y0 = yv.x, y1 = yv.y, y2 = yv.z, y3 = yv.w;
  const int yprev = __shfl_up(y3, 1, 32);        // label u-1 for this lane's k=0
  const bool al0 = (lane != 0) && (y0 != yprev); // s = 8*lane+1 (lane 0: s=1 -> false)
  const bool al1 = (y1 != y0);                   // s = 8*lane+3
  const bool al2 = (y2 != y1);                   // s = 8*lane+5
  const bool al3 = (y3 != y2);                   // s = 8*lane+7

  // ---- t = 0: initialize alpha from row 0 ----
  wait_async_15();                       // row 0 landed (oldest of 16 in flight)
  float a[9];
  {
    const float* st0 = rows;             // stage 0 holds row 0
    const float lpb0 = flog2(st0[BLANK] + EPSF);
    const float lpl0 = flog2(st0[y0] + EPSF);
    #pragma unroll
    for (int j = 0; j < 9; ++j) a[j] = NEGF;
    if (lane == 0) { a[0] = lpb0; a[1] = lpl0; }
  }
  async_row_load(yb + NS * C_ + lane * 4, rowsOff + laneB);   // refill stage 0 (row 16)

  // ---- peeled step t = 1 ----
  wait_async_15();
  ctc_step(a, rows + C_, y0, y1, y2, y3, al0, al1, al2, al3, lane);
  async_row_load(yb + (NS + 1) * C_ + lane * 4,
                 rowsOff + (uint32_t)(C_ * 4) + laneB);       // refill stage 1 (row 17)

  // ---- main scan, 2x software-pipelined: pairs (t, t+1), t = 2..1022 ----
  for (int t = 2; t < T_; t += 2) {
    const int s0i = t & (NS - 1);
    const int s1i = (t + 1) & (NS - 1);

    wait_async_14();                     // two oldest outstanding (rows t, t+1) landed

    // step t, then step t+1; the t+1 gathers/logs overlap the step-t lse chain
    ctc_step(a, rows + s0i * C_, y0, y1, y2, y3, al0, al1, al2, al3, lane);
    ctc_step(a, rows + s1i * C_, y0, y1, y2, y3, al0, al1, al2, al3, lane);

    // keep 16 ops in flight: refill both stages (clamped dummy rows near the
    // tail overwrite stages that are never read again)
    int r0 = t + NS;     if (r0 > T_ - 1) r0 = T_ - 1;
    int r1 = t + NS + 1; if (r1 > T_ - 1) r1 = T_ - 1;
    async_row_load(yb + r0 * C_ + lane * 4, rowsOff + (uint32_t)(s0i * C_ * 4) + laneB);
    async_row_load(yb + r1 * C_ + lane * 4, rowsOff + (uint32_t)(s1i * C_ * 4) + laneB);
  }

  // ---- epilogue: ll = ln2 * log2( 2^alpha[256] + 2^alpha[255] ) ----
  // state 256 = lane 31 j=8; state 255 = lane 31 j=7
  const float aS1 = __shfl(a[8], 31, 32);
  const float aS2 = __shfl(a[7], 31, 32);
  if (lane == 0) {
    const float ll = lse2(aS1, aS2) * 0.69314718055994531f;   // ln 2
    out[b] = -ll;
  }
  // S_ENDPGM's implicit wait-idle drains the remaining async ops.
}

extern "C" void kernel_launch(void* const* d_in, const int* in_sizes, int n_in,
                              void* d_out, int out_size, void* d_ws, size_t ws_size,
                              hipStream_t stream) {
  (void)in_sizes; (void)n_in; (void)out_size; (void)d_ws; (void)ws_size;
  const int*   y_true = (const int*)d_in[0];
  const float* y_pred = (const float*)d_in[1];
  float*       out    = (float*)d_out;
  hipLaunchKernelGGL(ctc_fwd_kernel, dim3(B_), dim3(32), 0, stream,
                     y_true, y_pred, out);
}